// CoreferenceSeparatedModel_79139067396559
// MI455X (gfx1250) — compile-verified
//
#include <hip/hip_runtime.h>
#include <hip/hip_bf16.h>
#include <math.h>

#define Bdim 2
#define Ldim 128
#define Hdim 768
#define Cdim 4
#define NEGV (-1024.0f)
#define MROWS (Bdim*Ldim)   // 256

typedef float v2f __attribute__((ext_vector_type(2)));
typedef float v8f __attribute__((ext_vector_type(8)));

// ---------------------------------------------------------------------------
// GEMM: out[M=256, N] = A[256, 768] @ W[768, N] + bias[N]
// One wave per 16x64 strip: 4 accumulators share one A fragment per K-step
// (V_WMMA_F32_16X16X4_F32).  A frag: lane l: M=l%16, K=2*(l/16)+{0,1}.
// B frag: N=l%16, same K.  C/D frag: elem v: M=v+8*(l/16), N=l%16.
// ---------------------------------------------------------------------------
__global__ __launch_bounds__(32) void gemm_f32_wmma(const float* __restrict__ A,
                                                    const float* __restrict__ W,
                                                    const float* __restrict__ bias,
                                                    float* __restrict__ out, int N) {
    const int K = Hdim;
    const int strips = N >> 6;                 // 64-wide strips
    const int tn = blockIdx.x % strips;
    const int tm = blockIdx.x / strips;
    const int lane = threadIdx.x;
    const int half = lane >> 4;
    const int lm   = lane & 15;

    const float* Arow = A + (size_t)(tm * 16 + lm) * K;
    const float* Wb   = W + (size_t)(tn * 64 + lm);

    v8f acc0 = {0.f,0.f,0.f,0.f,0.f,0.f,0.f,0.f};
    v8f acc1 = acc0, acc2 = acc0, acc3 = acc0;

    for (int k0 = 0; k0 < K; k0 += 4) {
        const int ka = k0 + 2 * half;
        v2f a;
        a.x = Arow[ka];
        a.y = Arow[ka + 1];
        const float* w0 = Wb + (size_t)ka * N;        // row k
        const float* w1 = Wb + (size_t)(ka + 1) * N;  // row k+1
        v2f b0, b1, b2, b3;
        b0.x = w0[0];  b0.y = w1[0];
        b1.x = w0[16]; b1.y = w1[16];
        b2.x = w0[32]; b2.y = w1[32];
        b3.x = w0[48]; b3.y = w1[48];
        acc0 = __builtin_amdgcn_wmma_f32_16x16x4_f32(false, a, false, b0, (short)0, acc0, false, false);
        acc1 = __builtin_amdgcn_wmma_f32_16x16x4_f32(false, a, false, b1, (short)0, acc1, false, false);
        acc2 = __builtin_amdgcn_wmma_f32_16x16x4_f32(false, a, false, b2, (short)0, acc2, false, false);
        acc3 = __builtin_amdgcn_wmma_f32_16x16x4_f32(false, a, false, b3, (short)0, acc3, false, false);
    }

    v8f* accs[4] = {&acc0, &acc1, &acc2, &acc3};
    #pragma unroll
    for (int t = 0; t < 4; ++t) {
        const int   col = tn * 64 + t * 16 + lm;
        const float bn  = bias[col];
        #pragma unroll
        for (int v = 0; v < 8; ++v) {
            const int row = tm * 16 + half * 8 + v;
            out[(size_t)row * N + col] = (*accs[t])[v] + bn;
        }
    }
}

// ---------------------------------------------------------------------------
// out_coref[b,i,j] = sum_h tanh(hp_c[b,i,h] + ha_c[b,j,h]) * wout_c[h] + mask
// One block per (b,i); 8 waves own j's, lanes split h.  Writes out_full[.,.,4,.]
// ---------------------------------------------------------------------------
__global__ __launch_bounds__(256) void coref_pairwise(const float* __restrict__ hp_c,
                                                      const float* __restrict__ ha_c,
                                                      const float* __restrict__ wout_c,
                                                      const unsigned char* __restrict__ ngmask,
                                                      const int* __restrict__ attn,
                                                      float* __restrict__ out_coref,
                                                      float* __restrict__ out_full) {
    __shared__ float s_hp[Hdim];
    __shared__ float s_w[Hdim];
    const int bi = blockIdx.x;          // b*L + i
    const int b  = bi / Ldim;
    const int tid = threadIdx.x;
    for (int h = tid; h < Hdim; h += 256) {
        s_hp[h] = hp_c[(size_t)bi * Hdim + h];
        s_w[h]  = wout_c[h];
    }
    __syncthreads();
    const int wave = tid >> 5, lane = tid & 31;
    for (int j = wave; j < Ldim; j += 8) {
        const float* ha = ha_c + (size_t)(b * Ldim + j) * Hdim;
        if (j + 8 < Ldim)   // cover L2 latency of the next streamed row
            __builtin_prefetch(ha_c + (size_t)(b * Ldim + j + 8) * Hdim + lane * 4, 0, 1);
        float acc = 0.f;
        for (int h = lane; h < Hdim; h += 32)
            acc += tanhf(s_hp[h] + ha[h]) * s_w[h];
        #pragma unroll
        for (int off = 16; off; off >>= 1) acc += __shfl_xor(acc, off, 32);
        if (lane == 0) {
            const bool m = (ngmask[((size_t)bi * 5 + 4) * Ldim + j] != 0) &&
                           (attn[b * Ldim + j] != 0);
            const float val = acc + (m ? 0.f : NEGV);
            out_coref[(size_t)bi * Ldim + j] = val;
            out_full[((size_t)bi * 5 + 4) * Ldim + j] = val;
        }
    }
}

// ---------------------------------------------------------------------------
// p = softmax(out_coref[b,i,:]); ctx[b,i,h] = sum_j p[j] * hidc[b,j,h]
// ---------------------------------------------------------------------------
__global__ __launch_bounds__(256) void softmax_context(const float* __restrict__ out_coref,
                                                       const float* __restrict__ hidc,
                                                       float* __restrict__ ctx) {
    __shared__ float sm[Ldim];
    __shared__ float s_max, s_inv;
    const int bi = blockIdx.x;
    const int b  = bi / Ldim;
    const int tid = threadIdx.x;
    if (tid < Ldim) sm[tid] = out_coref[(size_t)bi * Ldim + tid];
    __syncthreads();
    if (tid == 0) {
        float m = -INFINITY;
        for (int j = 0; j < Ldim; ++j) m = fmaxf(m, sm[j]);
        s_max = m;
    }
    __syncthreads();
    if (tid < Ldim) sm[tid] = expf(sm[tid] - s_max);
    __syncthreads();
    if (tid == 0) {
        float s = 0.f;
        for (int j = 0; j < Ldim; ++j) s += sm[j];
        s_inv = 1.0f / s;
    }
    __syncthreads();
    for (int h = tid; h < Hdim; h += 256) {
        float acc = 0.f;
        for (int j = 0; j < Ldim; ++j)
            acc += sm[j] * hidc[(size_t)(b * Ldim + j) * Hdim + h];
        ctx[(size_t)bi * Hdim + h] = acc * s_inv;
    }
}

// ---------------------------------------------------------------------------
// In-place fuse: ha0[b,j,c,h] += ctx[b,j,h]  (removes ctx stream + one add
// from the 101M-iteration pairwise inner loop)
// ---------------------------------------------------------------------------
__global__ __launch_bounds__(256) void fuse_ctx(float* __restrict__ ha0,
                                                const float* __restrict__ ctx) {
    const size_t idx = (size_t)blockIdx.x * 256 + threadIdx.x;  // < MROWS*C*H
    const int h  = (int)(idx % Hdim);
    const int bj = (int)(idx / ((size_t)Cdim * Hdim));
    ha0[idx] += ctx[(size_t)bj * Hdim + h];
}

// ---------------------------------------------------------------------------
// out_full[b,i,c,j] = sum_h tanh(hp[b,i,c,h] + haf[b,j,c,h]) * wout[h] + mask
// One block per (b,i); waves own (j,c) pairs, lanes split h.
// ---------------------------------------------------------------------------
__global__ __launch_bounds__(256) void main_pairwise(const float* __restrict__ hp,
                                                     const float* __restrict__ haf,
                                                     const float* __restrict__ wout,
                                                     const unsigned char* __restrict__ ngmask,
                                                     const int* __restrict__ attn,
                                                     float* __restrict__ out_full) {
    __shared__ float s_hp[Cdim * Hdim];   // 12 KB
    __shared__ float s_w[Hdim];
    const int bi = blockIdx.x;
    const int b  = bi / Ldim;
    const int tid = threadIdx.x;
    for (int h = tid; h < Cdim * Hdim; h += 256)
        s_hp[h] = hp[(size_t)bi * Cdim * Hdim + h];
    for (int h = tid; h < Hdim; h += 256) s_w[h] = wout[h];
    __syncthreads();
    const int wave = tid >> 5, lane = tid & 31;
    for (int p = wave; p < Cdim * Ldim; p += 8) {
        const int j = p >> 2, c = p & 3;
        const float* ha  = haf + ((size_t)(b * Ldim + j) * Cdim + c) * Hdim;
        const float* hpc = s_hp + c * Hdim;
        if (p + 8 < Cdim * Ldim) {
            const int jn = (p + 8) >> 2, cn = (p + 8) & 3;
            __builtin_prefetch(haf + ((size_t)(b * Ldim + jn) * Cdim + cn) * Hdim + lane * 4, 0, 1);
        }
        float acc = 0.f;
        for (int h = lane; h < Hdim; h += 32)
            acc += tanhf(hpc[h] + ha[h]) * s_w[h];
        #pragma unroll
        for (int off = 16; off; off >>= 1) acc += __shfl_xor(acc, off, 32);
        if (lane == 0) {
            const bool m = (ngmask[((size_t)bi * 5 + c) * Ldim + j] != 0) &&
                           (attn[b * Ldim + j] != 0);
            out_full[((size_t)bi * 5 + c) * Ldim + j] = acc + (m ? 0.f : NEGV);
        }
    }
}

// ---------------------------------------------------------------------------
// Per (b,i,c) row: CE partial = sum_j tgt*(logsumexp - logit); plus tgt sum.
// One wave per row, 4 waves per block.
// ---------------------------------------------------------------------------
__global__ __launch_bounds__(128) void loss_rows(const float* __restrict__ out_full,
                                                 const float* __restrict__ target,
                                                 float* __restrict__ loss_part,
                                                 float* __restrict__ tsum_part) {
    const int r    = blockIdx.x * 4 + (threadIdx.x >> 5);   // 0..1023
    const int lane = threadIdx.x & 31;
    const int bi = r >> 2, c = r & 3;
    const float* logit = out_full + ((size_t)bi * 5 + c) * Ldim;
    const float* tgt   = target   + ((size_t)bi * 5 + c) * Ldim;
    float v[4], mx = -INFINITY;
    #pragma unroll
    for (int q = 0; q < 4; ++q) { v[q] = logit[lane + 32 * q]; mx = fmaxf(mx, v[q]); }
    #pragma unroll
    for (int off = 16; off; off >>= 1) mx = fmaxf(mx, __shfl_xor(mx, off, 32));
    float se = 0.f;
    #pragma unroll
    for (int q = 0; q < 4; ++q) se += expf(v[q] - mx);
    #pragma unroll
    for (int off = 16; off; off >>= 1) se += __shfl_xor(se, off, 32);
    const float lse = logf(se) + mx;
    float lp = 0.f, ts = 0.f;
    #pragma unroll
    for (int q = 0; q < 4; ++q) {
        const float t = tgt[lane + 32 * q];
        lp += t * (lse - v[q]);
        ts += t;
    }
    #pragma unroll
    for (int off = 16; off; off >>= 1) {
        lp += __shfl_xor(lp, off, 32);
        ts += __shfl_xor(ts, off, 32);
    }
    if (lane == 0) { loss_part[r] = lp; tsum_part[r] = ts; }
}

// Deterministic final reduction of 1024 partials -> d_out[0]
__global__ __launch_bounds__(256) void finalize_loss(const float* __restrict__ loss_part,
                                                     const float* __restrict__ tsum_part,
                                                     float* __restrict__ d_out) {
    __shared__ float sl[256], st[256];
    const int tid = threadIdx.x;
    float a = 0.f, t = 0.f;
    #pragma unroll
    for (int q = 0; q < 4; ++q) { a += loss_part[tid * 4 + q]; t += tsum_part[tid * 4 + q]; }
    sl[tid] = a; st[tid] = t;
    __syncthreads();
    for (int s = 128; s; s >>= 1) {
        if (tid < s) { sl[tid] += sl[tid + s]; st[tid] += st[tid + s]; }
        __syncthreads();
    }
    if (tid == 0) d_out[0] = sl[0] / fmaxf(st[0], 1.0f);
}

extern "C" void kernel_launch(void* const* d_in, const int* in_sizes, int n_in,
                              void* d_out, int out_size, void* d_ws, size_t ws_size,
                              hipStream_t stream) {
    const float* seq      = (const float*)d_in[0];
    const float* seq_c    = (const float*)d_in[1];
    const int*   attn     = (const int*)d_in[2];
    const unsigned char* ngm = (const unsigned char*)d_in[3];   // jax bool -> 1 byte
    const float* target   = (const float*)d_in[4];
    const float* Wc   = (const float*)d_in[5];
    const float* bc   = (const float*)d_in[6];
    const float* Wp   = (const float*)d_in[7];
    const float* bp   = (const float*)d_in[8];
    const float* Wa   = (const float*)d_in[9];
    const float* ba   = (const float*)d_in[10];
    const float* wout = (const float*)d_in[11];
    const float* Wp_c = (const float*)d_in[12];
    const float* bp_c = (const float*)d_in[13];
    const float* Wa_c = (const float*)d_in[14];
    const float* ba_c = (const float*)d_in[15];
    const float* wout_c = (const float*)d_in[16];

    float* outp     = (float*)d_out;        // [0]=loss, [1..]=out_full (B,L,5,L)
    float* out_full = outp + 1;

    // workspace carve-up (floats) -- ~9.6 MB
    float* ws = (float*)d_ws;
    size_t o = 0;
    float* hp_c  = ws + o; o += (size_t)MROWS * Hdim;            // 256x768
    float* ha_c  = ws + o; o += (size_t)MROWS * Hdim;
    float* hidc  = ws + o; o += (size_t)MROWS * Hdim;
    float* hp    = ws + o; o += (size_t)MROWS * Cdim * Hdim;     // 256x3072
    float* ha0   = ws + o; o += (size_t)MROWS * Cdim * Hdim;     // fused in place
    float* ocref = ws + o; o += (size_t)MROWS * Ldim;            // 256x128
    float* ctx   = ws + o; o += (size_t)MROWS * Hdim;
    float* lpart = ws + o; o += 1024;
    float* tpart = ws + o; o += 1024;

    const dim3 wblk(32);
    const int strips768  = (MROWS / 16) * (Hdim / 64);            // 16*12
    const int strips3072 = (MROWS / 16) * ((Cdim * Hdim) / 64);   // 16*48

    // Projections (WMMA f32 16x16x4, 16x64 strip per wave)
    gemm_f32_wmma<<<strips768,  wblk, 0, stream>>>(seq_c, Wp_c, bp_c, hp_c, Hdim);
    gemm_f32_wmma<<<strips768,  wblk, 0, stream>>>(seq_c, Wa_c, ba_c, ha_c, Hdim);
    gemm_f32_wmma<<<strips768,  wblk, 0, stream>>>(seq,   Wc,   bc,   hidc, Hdim);
    gemm_f32_wmma<<<strips3072, wblk, 0, stream>>>(seq,   Wp,   bp,   hp,   Cdim * Hdim);
    gemm_f32_wmma<<<strips3072, wblk, 0, stream>>>(seq,   Wa,   ba,   ha0,  Cdim * Hdim);

    // Coref pairwise scores (+ out_full[:,:,4,:])
    coref_pairwise<<<MROWS, 256, 0, stream>>>(hp_c, ha_c, wout_c, ngm, attn, ocref, out_full);

    // Softmax over coref scores + context vectors
    softmax_context<<<MROWS, 256, 0, stream>>>(ocref, hidc, ctx);

    // Fold context into ha0 (in place)
    fuse_ctx<<<(MROWS * Cdim * Hdim) / 256, 256, 0, stream>>>(ha0, ctx);

    // Main pairwise scores (+ out_full[:,:,0:4,:])
    main_pairwise<<<MROWS, 256, 0, stream>>>(hp, ha0, wout, ngm, attn, out_full);

    // Cross-entropy loss
    loss_rows<<<(MROWS * Cdim) / 4, 128, 0, stream>>>(out_full, target, lpart, tpart);
    finalize_loss<<<1, 256, 0, stream>>>(lpart, tpart, outp);
}